// SimplifiedPrecisionAttentionBlock_44873818309052
// MI455X (gfx1250) — compile-verified
//
#include <hip/hip_runtime.h>
#include <math.h>

#define Bb 4
#define Nn 2048
#define Dd 256

typedef __attribute__((ext_vector_type(2))) float v2f;
typedef __attribute__((ext_vector_type(8))) float v8f;

#define QS_STRIDE 260
#define SB_STRIDE 2052
#define NEG_INF (-__builtin_inff())

// ---------------------------------------------------------------------------
// Complex-linear projection: Out[b,m,n] = sum_k Zr[b,m,k]*Wa[n,k]
//                                       + sum_k Zi[b,m,k]*Wb[n,k] + bias[n]
// (sign of the second term is pre-folded into Wb by negate_kernel)
// One wave computes one 16x16 output tile via V_WMMA_F32_16X16X4_F32, K=256.
// ---------------------------------------------------------------------------
__global__ void clinear_wmma(const float* __restrict__ Zr, const float* __restrict__ Zi,
                             long zStride,
                             const float* __restrict__ Wa, const float* __restrict__ Wb,
                             const float* __restrict__ bias,
                             float* __restrict__ Out, long oStride)
{
    const int lane  = threadIdx.x & 31;
    const int wave  = threadIdx.x >> 5;
    const int tilesN = Dd / 16;                 // 16
    const int tilesM = Nn / 16;                 // 128
    const int perBatch = tilesM * tilesN;       // 2048

    int tile = blockIdx.x * 8 + wave;           // 8192 tiles total
    int b  = tile / perBatch;
    int t  = tile - b * perBatch;
    int mT = t / tilesN;
    int nT = t - mT * tilesN;

    const int l15   = lane & 15;
    const int khalf = (lane >> 4) << 1;
    const int ncol  = nT * 16 + l15;

    const float* zr = Zr + (long)b * zStride + (long)(mT * 16 + l15) * Dd;
    const float* zi = Zi + (long)b * zStride + (long)(mT * 16 + l15) * Dd;
    const float* wa = Wa + (long)ncol * Dd;
    const float* wb = Wb + (long)ncol * Dd;

    v8f acc = {};
    for (int k = 0; k < Dd; k += 4) {
        v2f a, w;
        a.x = zr[k + khalf];
        a.y = zr[k + khalf + 1];
        w.x = wa[k + khalf];
        w.y = wa[k + khalf + 1];
        acc = __builtin_amdgcn_wmma_f32_16x16x4_f32(false, a, false, w,
                                                    (short)0, acc, false, false);
        a.x = zi[k + khalf];
        a.y = zi[k + khalf + 1];
        w.x = wb[k + khalf];
        w.y = wb[k + khalf + 1];
        acc = __builtin_amdgcn_wmma_f32_16x16x4_f32(false, a, false, w,
                                                    (short)0, acc, false, false);
    }

    float bv = bias[ncol];
    float* outp = Out + (long)b * oStride;
    const int rbase = (lane >> 4) << 3;
#pragma unroll
    for (int r = 0; r < 8; ++r) {
        int row = mT * 16 + rbase + r;
        outp[(long)row * Dd + ncol] = acc[r] + bv;
    }
}

// ---------------------------------------------------------------------------
__global__ void negate_kernel(float* __restrict__ dst, const float* __restrict__ src, long n)
{
    long i = (long)blockIdx.x * blockDim.x + threadIdx.x;
    long stride = (long)gridDim.x * blockDim.x;
    for (; i < n; i += stride) dst[i] = -src[i];
}

// ---------------------------------------------------------------------------
// Row squared-norms of Q_r and K_r (one wave per row).
// ---------------------------------------------------------------------------
__global__ void row_norms(const float* __restrict__ Qr, const float* __restrict__ Kr,
                          float* __restrict__ qn, float* __restrict__ kn)
{
    int lane = threadIdx.x & 31;
    int wave = threadIdx.x >> 5;
    int row  = blockIdx.x * 8 + wave;            // 0 .. 2*B*N-1
    const float* src; float* dst; int r;
    if (row < Bb * Nn) { src = Qr; dst = qn; r = row; }
    else               { src = Kr; dst = kn; r = row - Bb * Nn; }
    const float* p = src + (long)r * Dd;
    float s = 0.f;
    for (int d = lane; d < Dd; d += 32) { float v = p[d]; s += v * v; }
#pragma unroll
    for (int off = 16; off > 0; off >>= 1) s += __shfl_xor(s, off, 32);
    if (lane == 0) dst[r] = s;
}

// ---------------------------------------------------------------------------
// Scalar reductions: ga_bar, om_bar, tau^2, sigmoid(delta).
// ---------------------------------------------------------------------------
__global__ void scalars_kernel(const float* __restrict__ lamOm, const float* __restrict__ lamGa,
                               const float* __restrict__ lamC,  const float* __restrict__ tau,
                               const float* __restrict__ delta, float* __restrict__ scal)
{
    __shared__ float red[256];
    int t = threadIdx.x;
    float c2 = lamC[t] * lamC[t];
    float ga = lamGa[t] * lamGa[t] + 1e-5f;
    float om = lamOm[t] * lamOm[t] + 1e-5f;

    red[t] = c2 * ga;
    __syncthreads();
    for (int s = 128; s > 0; s >>= 1) { if (t < s) red[t] += red[t + s]; __syncthreads(); }
    float gaBar = red[0] * (1.0f / Dd);
    __syncthreads();

    red[t] = c2 * om;
    __syncthreads();
    for (int s = 128; s > 0; s >>= 1) { if (t < s) red[t] += red[t + s]; __syncthreads(); }
    if (t == 0) {
        scal[0] = gaBar;
        scal[1] = red[0] * (1.0f / Dd);
        scal[2] = tau[0] * tau[0];
        scal[3] = 1.0f / (1.0f + __expf(-delta[0]));
    }
}

// ---------------------------------------------------------------------------
// Fused attention band: one block owns 16 query rows of one batch.
// Score band (16 x 2048) lives entirely in LDS (CDNA5: 320KB/WGP).
// Phase 0: Q band staged via global_load_async_to_lds_b128 (ASYNCcnt path)
// Phase 1: S = -tau^2*log(Vavg + |q|^2 + |k|^2 - 2 QK^T)   (WMMA f32)
// Phase 2: causal softmax in LDS (A + explicit zero tail kept in LDS)
// Phase 2b: Q_ij real plane streamed out via global_store_async_from_lds_b128
// Phase 3: est = A @ V_r (WMMA f32, K limited to causal extent)
// ---------------------------------------------------------------------------
__global__ void attn_band(const float* __restrict__ Qr, const float* __restrict__ Kr,
                          const float* __restrict__ Vr,
                          const float* __restrict__ qn, const float* __restrict__ kn,
                          const float* __restrict__ tvec, const float* __restrict__ scal,
                          float* __restrict__ outQij, float* __restrict__ outE)
{
    extern __shared__ float smem[];
    float* qs = smem;                       // 16 * QS_STRIDE
    float* sb = smem + 16 * QS_STRIDE;      // 16 * SB_STRIDE
    float* red = sb + 16 * SB_STRIDE;       // 256

    const int tid  = threadIdx.x;
    const int lane = tid & 31;
    const int wave = tid >> 5;
    const int tilesI = Nn / 16;             // 128
    const int b  = blockIdx.x / tilesI;
    const int iT = blockIdx.x - b * tilesI;
    const int iBase = iT * 16;

    const float gaBar = scal[0], omBar = scal[1], tau2 = scal[2], sMix = scal[3];

    // ---- Phase 0: async DMA of the Q band into LDS (16B chunks, aligned) ----
    const float* qsrc = Qr + ((long)b * Nn + iBase) * Dd;
    for (int c = tid; c < (16 * Dd) / 4; c += 256) {     // 1024 chunks of 16B
        int rr = c >> 6;              // (c*4)/Dd
        int cc = (c & 63) << 2;       // (c*4)%Dd
        unsigned ldsOff = (unsigned)(unsigned long long)(qs + rr * QS_STRIDE + cc);
        unsigned long long ga = (unsigned long long)(qsrc + (long)c * 4);
        asm volatile("global_load_async_to_lds_b128 %0, %1, off"
                     :: "v"(ldsOff), "v"(ga) : "memory");
    }
    asm volatile("s_wait_asynccnt 0" ::: "memory");
    __syncthreads();

    const int numJT = iT + 1;
    const int L = numJT * 16;       // causal extent
    const int l15   = lane & 15;
    const int khalf = (lane >> 4) << 1;
    const int rbase = (lane >> 4) << 3;

    // ---- Phase 1: score tiles (wave-cyclic over j-tiles) ----
    for (int jt = wave; jt < numJT; jt += 8) {
        const int jBase = jt * 16;
        const float* kp = Kr + ((long)b * Nn + jBase + l15) * Dd;
        if (jt + 8 < numJT)
            __builtin_prefetch(kp + 8 * 16 * Dd, 0, 1);
        v8f acc = {};
        for (int k = 0; k < Dd; k += 4) {
            v2f a, w;
            a.x = qs[l15 * QS_STRIDE + k + khalf];
            a.y = qs[l15 * QS_STRIDE + k + khalf + 1];
            w.x = kp[k + khalf];
            w.y = kp[k + khalf + 1];
            acc = __builtin_amdgcn_wmma_f32_16x16x4_f32(false, a, false, w,
                                                        (short)0, acc, false, false);
        }
        const int j  = jBase + l15;
        const float knj = kn[b * Nn + j];
        const float tj  = tvec[j];
#pragma unroll
        for (int r = 0; r < 8; ++r) {
            int rr = rbase + r;
            int i  = iBase + rr;
            float R    = qn[b * Nn + i] + knj - 2.0f * acc[r];
            float Vavg = gaBar + omBar * fabsf(tvec[i] - tj);
            float sc   = -tau2 * __logf(Vavg + R);
            if (j > i) sc = NEG_INF;
            sb[rr * SB_STRIDE + j] = sc;
        }
    }
    __syncthreads();

    // ---- Phase 2: causal softmax (16 threads per row), A kept in LDS ----
    {
        const int row = tid >> 4;
        const int sub = tid & 15;
        float m = NEG_INF;
        for (int c = sub; c < L; c += 16) m = fmaxf(m, sb[row * SB_STRIDE + c]);
        red[tid] = m;
        __syncthreads();
        float rm = NEG_INF;
#pragma unroll
        for (int u = 0; u < 16; ++u) rm = fmaxf(rm, red[row * 16 + u]);
        __syncthreads();
        float ssum = 0.f;
        for (int c = sub; c < L; c += 16) {
            float e = __expf(sb[row * SB_STRIDE + c] - rm);
            sb[row * SB_STRIDE + c] = e;
            ssum += e;
        }
        red[tid] = ssum;
        __syncthreads();
        float rs = 0.f;
#pragma unroll
        for (int u = 0; u < 16; ++u) rs += red[row * 16 + u];
        float inv = 1.0f / rs;
        for (int c = sub; c < Nn; c += 16) {
            float av = 0.f;
            if (c < L) av = sb[row * SB_STRIDE + c] * inv;
            sb[row * SB_STRIDE + c] = av;       // masked tail -> explicit zeros
        }
    }
    __syncthreads();

    // ---- Phase 2b: stream Q_ij real band out via LDS->global async DMA ----
    {
        float* qbase = outQij + ((long)(b * 2) * Nn + iBase) * Nn;
        for (int c = tid; c < 16 * (Nn / 4); c += 256) {  // 8192 chunks of 16B
            int row = c >> 9;
            int col = (c & 511) << 2;
            unsigned ldsOff = (unsigned)(unsigned long long)(sb + row * SB_STRIDE + col);
            unsigned long long ga = (unsigned long long)(qbase + (long)row * Nn + col);
            asm volatile("global_store_async_from_lds_b128 %0, %1, off"
                         :: "v"(ga), "v"(ldsOff) : "memory");
        }
        // completion covered by implicit S_WAIT_IDLE at S_ENDPGM; phase 3 only
        // reads the same LDS region, so no ordering hazard.
    }

    // ---- Phase 3: est = A @ V_r over causal extent; real eigenbasis out ----
    for (int nt = wave; nt < 16; nt += 8) {
        const int dcol = nt * 16 + l15;
        v8f acc = {};
        for (int k = 0; k < L; k += 4) {
            v2f a, w;
            a.x = sb[l15 * SB_STRIDE + k + khalf];
            a.y = sb[l15 * SB_STRIDE + k + khalf + 1];
            const float* vp = Vr + ((long)b * Nn + k + khalf) * Dd + dcol;
            w.x = vp[0];
            w.y = vp[Dd];
            acc = __builtin_amdgcn_wmma_f32_16x16x4_f32(false, a, false, w,
                                                        (short)0, acc, false, false);
        }
#pragma unroll
        for (int r = 0; r < 8; ++r) {
            int rr = rbase + r;
            int i  = iBase + rr;
            long gi = ((long)b * Nn + i) * Dd + dcol;
            float er = (1.0f - sMix) * Vr[gi] + sMix * acc[r];
            outE[((long)(b * 2) * Nn + i) * Dd + dcol] = er;
        }
    }
}

// ---------------------------------------------------------------------------
// Epilogue kernels
// ---------------------------------------------------------------------------
__global__ void fill_kernel(float* __restrict__ p, float v, long n)
{
    long i = (long)blockIdx.x * blockDim.x + threadIdx.x;
    long stride = (long)gridDim.x * blockDim.x;
    for (; i < n; i += stride) p[i] = v;
}

__global__ void eig_imag_kernel(float* __restrict__ outE, const float* __restrict__ Vi,
                                const float* __restrict__ scal)
{
    const float s1 = 1.0f - scal[3];
    const long n = (long)Bb * Nn * Dd;
    const long plane = (long)Nn * Dd;
    long i = (long)blockIdx.x * blockDim.x + threadIdx.x;
    long stride = (long)gridDim.x * blockDim.x;
    for (; i < n; i += stride) {
        long b = i / plane;
        long rem = i - b * plane;
        outE[(2 * b + 1) * plane + rem] = s1 * Vi[i];
    }
}

__global__ void uv_kernel(float* __restrict__ outU, const float* __restrict__ Vr)
{
    const long n = (long)Bb * Nn * Dd;
    const long plane = (long)Nn * Dd;
    long i = (long)blockIdx.x * blockDim.x + threadIdx.x;
    long stride = (long)gridDim.x * blockDim.x;
    for (; i < n; i += stride) {
        long b = i / plane;
        long rem = i - b * plane;
        outU[(2 * b) * plane + rem]     = Vr[i];   // U_v real = V_r
        outU[(2 * b + 1) * plane + rem] = 0.f;     // U_v imag = 0
    }
}

__global__ void qij_imag_kernel(float* __restrict__ q)
{
    const long plane = (long)Nn * Nn;
    const long n = (long)Bb * plane;
    long i = (long)blockIdx.x * blockDim.x + threadIdx.x;
    long stride = (long)gridDim.x * blockDim.x;
    for (; i < n; i += stride) {
        long b = i / plane;
        long rem = i - b * plane;
        q[(2 * b + 1) * plane + rem] = 0.f;
    }
}

// ---------------------------------------------------------------------------
extern "C" void kernel_launch(void* const* d_in, const int* in_sizes, int n_in,
                              void* d_out, int out_size, void* d_ws, size_t ws_size,
                              hipStream_t stream)
{
    const float* Zq    = (const float*)d_in[0];
    const float* Zk    = (const float*)d_in[1];
    const float* Zv    = (const float*)d_in[2];
    const float* tvec  = (const float*)d_in[3];   // row 0, first N entries
    const float* Wq    = (const float*)d_in[4];
    const float* bq    = (const float*)d_in[5];
    const float* Wk    = (const float*)d_in[6];
    const float* bk    = (const float*)d_in[7];
    const float* Wv    = (const float*)d_in[8];
    const float* bv    = (const float*)d_in[9];
    const float* Wp    = (const float*)d_in[10];
    const float* bp    = (const float*)d_in[11];
    const float* lamOm = (const float*)d_in[13];
    const float* lamGa = (const float*)d_in[14];
    const float* tau   = (const float*)d_in[15];
    const float* delta = (const float*)d_in[16];
    const float* lamC  = (const float*)d_in[17];

    const long ND  = (long)Nn * Dd;        // 524288
    const long BND = (long)Bb * ND;        // 2097152
    const long DD  = (long)Dd * Dd;        // 65536

    float* ws   = (float*)d_ws;
    float* scal = ws;                      // 16 floats
    float* Qr   = ws + 16;
    float* Kr   = Qr + BND;
    float* Vr   = Kr + BND;
    float* Vi   = Vr + BND;
    float* qn   = Vi + BND;
    float* kn   = qn + (long)Bb * Nn;
    float* nWq1 = kn + (long)Bb * Nn;      // negated imaginary weights
    float* nWk1 = nWq1 + DD;
    float* nWv1 = nWk1 + DD;
    float* nWp1 = nWv1 + DD;

    float* out   = (float*)d_out;
    float* outE  = out;                                // est_eigenbasis (B,2,N,D)
    float* outO  = outE + 2 * BND;                     // out            (B,2,N,D)
    float* outQ  = outO + 2 * BND;                     // Q_ij           (B,2,N,N)
    float* outVv = outQ + (long)Bb * 2 * Nn * Nn;      // V_v            (1,2,N,D)
    float* outUv = outVv + 2 * ND;                     // U_v            (B,2,N,D)
    float* outLh = outUv + 2 * BND;                    // lambda_h       (2,D,1)

    // scalar reductions + weight pre-negation (folds the complex sign)
    scalars_kernel<<<1, 256, 0, stream>>>(lamOm, lamGa, lamC, tau, delta, scal);
    negate_kernel<<<64, 256, 0, stream>>>(nWq1, Wq + DD, DD);
    negate_kernel<<<64, 256, 0, stream>>>(nWk1, Wk + DD, DD);
    negate_kernel<<<64, 256, 0, stream>>>(nWv1, Wv + DD, DD);
    negate_kernel<<<64, 256, 0, stream>>>(nWp1, Wp + DD, DD);

    // complex-linear projections (WMMA f32): Q_r, K_r, V_r, V_i
    clinear_wmma<<<1024, 256, 0, stream>>>(Zq, Zq + ND, 2 * ND, Wq, nWq1, bq,      Qr, ND);
    clinear_wmma<<<1024, 256, 0, stream>>>(Zk, Zk + ND, 2 * ND, Wk, nWk1, bk,      Kr, ND);
    clinear_wmma<<<1024, 256, 0, stream>>>(Zv, Zv + ND, 2 * ND, Wv, nWv1, bv,      Vr, ND);
    clinear_wmma<<<1024, 256, 0, stream>>>(Zv, Zv + ND, 2 * ND, Wv + DD, Wv, bv + Dd, Vi, ND);

    // row norms of Q_r / K_r
    row_norms<<<2048, 256, 0, stream>>>(Qr, Kr, qn, kn);

    // est_eigenbasis imag = (1-s) * V_i
    eig_imag_kernel<<<2048, 256, 0, stream>>>(outE, Vi, scal);

    // fused attention bands (dynamic LDS: Q band + score band + reduce scratch)
    const int smemBytes = (16 * QS_STRIDE + 16 * SB_STRIDE + 256) * sizeof(float);
    attn_band<<<Bb * (Nn / 16), 256, smemBytes, stream>>>(Qr, Kr, Vr, qn, kn,
                                                          tvec, scal, outQ, outE);

    // output projection: out = clinear(est_eigenbasis, W_p, b_p)
    clinear_wmma<<<1024, 256, 0, stream>>>(outE, outE + ND, 2 * ND, Wp, nWp1, bp,      outO,      2 * ND);
    clinear_wmma<<<1024, 256, 0, stream>>>(outE, outE + ND, 2 * ND, Wp + DD, Wp, bp + Dd, outO + ND, 2 * ND);

    // constant / copy outputs
    fill_kernel<<<512, 256, 0, stream>>>(outVv, 1.0f, ND);            // V_v real = 1
    fill_kernel<<<512, 256, 0, stream>>>(outVv + ND, 0.0f, ND);       // V_v imag = 0
    uv_kernel<<<2048, 256, 0, stream>>>(outUv, Vr);                   // U_v = [V_r, 0]
    qij_imag_kernel<<<8192, 256, 0, stream>>>(outQ);                  // Q_ij imag = 0
    fill_kernel<<<1, 256, 0, stream>>>(outLh, 0.0f, 2 * Dd);          // lambda_h = 0
}